// DistSparseMoe_28638841930253
// MI455X (gfx1250) — compile-verified
//
#include <hip/hip_runtime.h>
#include <hip/hip_bf16.h>

typedef __attribute__((ext_vector_type(2))) float v2f;
typedef __attribute__((ext_vector_type(8))) float v8f;

#define TOK   16384
#define HDIM  1024
#define ENUM  8
#define CAP   100
#define ECS   800      // ENUM * CAP
#define KC    256      // K chunk staged in LDS
#define NT    8        // N tiles (of 16) per wave

// ---------------- Router: 1 wave per token ----------------
__global__ __launch_bounds__(256) void k_router(const float* __restrict__ x,
                                                const float* __restrict__ wg,
                                                float* __restrict__ gate,
                                                int* __restrict__ best) {
    const int lane = threadIdx.x & 31;
    const int wave = threadIdx.x >> 5;
    const int t = blockIdx.x * 8 + wave;
    const float* xr = x + (size_t)t * HDIM;

    float acc[ENUM];
#pragma unroll
    for (int e = 0; e < ENUM; ++e) acc[e] = 0.f;

#pragma unroll 4
    for (int i = 0; i < HDIM / 32; ++i) {
        const int h = i * 32 + lane;
        const float xv = xr[h];                       // coalesced 128B per wave
        const float4 g0 = *(const float4*)(wg + h * ENUM);
        const float4 g1 = *(const float4*)(wg + h * ENUM + 4);
        acc[0] += xv * g0.x; acc[1] += xv * g0.y;
        acc[2] += xv * g0.z; acc[3] += xv * g0.w;
        acc[4] += xv * g1.x; acc[5] += xv * g1.y;
        acc[6] += xv * g1.z; acc[7] += xv * g1.w;
    }
#pragma unroll
    for (int off = 16; off > 0; off >>= 1) {
#pragma unroll
        for (int e = 0; e < ENUM; ++e)
            acc[e] += __shfl_xor(acc[e], off, 32);
    }
    if (lane == 0) {
        float mx = acc[0]; int bi = 0;
#pragma unroll
        for (int e = 1; e < ENUM; ++e)
            if (acc[e] > mx) { mx = acc[e]; bi = e; }   // first-max, matches jnp.argmax
        float s = 0.f;
#pragma unroll
        for (int e = 0; e < ENUM; ++e) s += expf(acc[e] - mx);
        gate[t] = 1.0f / s;   // softmax prob of the argmax
        best[t] = bi;
    }
}

// ---------------- Slot assignment: single wave ballot-scan ----------------
__global__ void k_scan(const int* __restrict__ best,
                       int* __restrict__ slotOf,
                       int* __restrict__ tokOf) {
    const int lane = threadIdx.x;                     // 32 threads
    for (int s = lane; s < ECS; s += 32) tokOf[s] = -1;

    int base[ENUM];
#pragma unroll
    for (int e = 0; e < ENUM; ++e) base[e] = 0;
    const unsigned lmask = (lane == 0) ? 0u : (0xFFFFFFFFu >> (32 - lane));

    for (int t0 = 0; t0 < TOK; t0 += 32) {
        const int b = best[t0 + lane];
        int myslot = -1;
#pragma unroll
        for (int e = 0; e < ENUM; ++e) {
            const unsigned m = (unsigned)__ballot(b == e);
            if (b == e) {
                const int c = base[e] + __popc(m & lmask);
                if (c < CAP) myslot = e * CAP + c;
            }
            base[e] += __popc(m);
        }
        slotOf[t0 + lane] = myslot;
        if (myslot >= 0) tokOf[myslot] = t0 + lane;
    }
}

// ---------------- Expert GEMM: (800x1024) @ (1024x1024) + bias via WMMA f32 ----------------
__global__ __launch_bounds__(256) void k_gemm(const float* __restrict__ x,
                                              const float* __restrict__ w,
                                              const float* __restrict__ bias,
                                              const int* __restrict__ tokOf,
                                              float* __restrict__ eo) {
    __shared__ float As[16][KC + 4];                  // +4 pad: conflict-free f32 frag reads

    const int tid  = threadIdx.x;
    const int lane = tid & 31;
    const int wave = tid >> 5;
    const int mbase = blockIdx.x * 16;                // 50 blocks * 16 = 800 exactly

    // A staging assignment: 16 threads per row, 16 floats (4x float4) each
    const int arow  = tid >> 4;                       // 0..15
    const int acol0 = (tid & 15) * 16;
    const int tok   = tokOf[mbase + arow];            // -1 => zero row

    v8f acc[NT];
    const v8f vzero = {0.f,0.f,0.f,0.f,0.f,0.f,0.f,0.f};
#pragma unroll
    for (int j = 0; j < NT; ++j) acc[j] = vzero;

    const int koff = (lane >> 4) << 1;                // 0 or 2 (A/B fragment K-offset)
    const int m    = lane & 15;
    const int n15  = lane & 15;

    for (int kc = 0; kc < HDIM; kc += KC) {
        __syncthreads();
        if (tok >= 0) {
            const float* src = x + (size_t)tok * HDIM + kc + acol0;
#pragma unroll
            for (int q = 0; q < 16; q += 4)
                *(float4*)&As[arow][acol0 + q] = *(const float4*)(src + q);
        } else {
            const float4 z4 = {0.f, 0.f, 0.f, 0.f};
#pragma unroll
            for (int q = 0; q < 16; q += 4)
                *(float4*)&As[arow][acol0 + q] = z4;
        }
        __syncthreads();

#pragma unroll
        for (int j = 0; j < NT; ++j) {
            const int ncol = (wave * NT + j) * 16 + n15;
            const float* wp = w + (size_t)(kc + koff) * HDIM + ncol;
#pragma unroll 8
            for (int k = 0; k < KC; k += 4) {
                // A fragment (16x4 f32): lanes 0-15 hold K=koff,koff+1; lanes 16-31 K=koff+2,koff+3
                v2f a = *(const v2f*)&As[m][k + koff];
                // B fragment (4x16 f32): mirrored layout, per-lane column n
                v2f b;
                b.x = wp[(size_t)k * HDIM];
                b.y = wp[(size_t)(k + 1) * HDIM];
                acc[j] = __builtin_amdgcn_wmma_f32_16x16x4_f32(
                    false, a, false, b, (short)0, acc[j], false, false);
            }
        }
    }

    // C/D layout: VGPR r -> M=r (lanes 0-15) / M=r+8 (lanes 16-31), N = lane%16
    const int rowoff = (lane >> 4) * 8;
#pragma unroll
    for (int j = 0; j < NT; ++j) {
        const int ncol = (wave * NT + j) * 16 + n15;
        const float bv = bias[ncol];
#pragma unroll
        for (int r = 0; r < 8; ++r)
            eo[(size_t)(mbase + rowoff + r) * HDIM + ncol] = acc[j][r] + bv;
    }
}

// ---------------- Combine: out[t] = gate[t] * eo[slot[t]] (0 if dropped) ----------------
__global__ __launch_bounds__(256) void k_combine(const float* __restrict__ eo,
                                                 const float* __restrict__ gate,
                                                 const int* __restrict__ slotOf,
                                                 float* __restrict__ out) {
    const int t  = blockIdx.x;
    const int s  = slotOf[t];
    const int c0 = threadIdx.x * 4;
    float4 r = {0.f, 0.f, 0.f, 0.f};
    if (s >= 0) {
        const float g = gate[t];
        const float4 v = *(const float4*)(eo + (size_t)s * HDIM + c0);
        r.x = g * v.x; r.y = g * v.y; r.z = g * v.z; r.w = g * v.w;
    }
    *(float4*)(out + (size_t)t * HDIM + c0) = r;
}

extern "C" void kernel_launch(void* const* d_in, const int* in_sizes, int n_in,
                              void* d_out, int out_size, void* d_ws, size_t ws_size,
                              hipStream_t stream) {
    (void)in_sizes; (void)n_in; (void)out_size; (void)ws_size;
    const float* x  = (const float*)d_in[0];   // (8,2048,1024)
    const float* wg = (const float*)d_in[1];   // (1024,8)
    const float* we = (const float*)d_in[2];   // (1024,1024)
    const float* be = (const float*)d_in[3];   // (1024,)
    float* out = (float*)d_out;

    char* ws = (char*)d_ws;
    float* gate  = (float*)(ws + 0);           // 16384 f32
    int*   best  = (int*)(ws + 65536);         // 16384 i32
    int*   slotOf= (int*)(ws + 131072);        // 16384 i32
    int*   tokOf = (int*)(ws + 196608);        // 800 i32
    float* eo    = (float*)(ws + 200704);      // 800*1024 f32

    k_router <<<dim3(TOK / 8), 256, 0, stream>>>(x, wg, gate, best);
    k_scan   <<<dim3(1),        32, 0, stream>>>(best, slotOf, tokOf);
    k_gemm   <<<dim3(ECS / 16),256, 0, stream>>>(x, we, be, tokOf, eo);
    k_combine<<<dim3(TOK),     256, 0, stream>>>(eo, gate, slotOf, out);
}